// LogicAwareConv_35897336660715
// MI455X (gfx1250) — compile-verified
//
#include <hip/hip_runtime.h>

typedef __attribute__((ext_vector_type(2))) float v2f;
typedef __attribute__((ext_vector_type(8))) float v8f;

// ---------------------------------------------------------------------------
// Cl(3,0) structure constants baked as WMMA B-operand.
// Blade order: ['', e1, e2, e3, e12, e13, e23, e123]
// B matrix: rows = pair index c = i*8+j (K=64), cols = output blade k (N=16,
// cols 8..15 zero padding). Exactly one +-1 per row.
// ---------------------------------------------------------------------------
struct BTab { float v[64 * 16]; };
struct CTab { int tgt[64]; float sgn[64]; };

static constexpr int MASKS_[8] = {0,1,2,4,3,5,6,7};
static constexpr int IDX_[8]   = {0,1,2,4,3,5,6,7}; // mask -> blade index

static constexpr float reorder_sign_(int a, int b) {
  int aa = a >> 1, s = 0;
  while (aa) { int m = aa & b; while (m) { s += m & 1; m >>= 1; } aa >>= 1; }
  return (s & 1) ? -1.0f : 1.0f;
}
static constexpr BTab make_btab() {
  BTab t{};
  for (int i = 0; i < 8; ++i)
    for (int j = 0; j < 8; ++j) {
      int a = MASKS_[i], b = MASKS_[j];
      t.v[(i*8 + j)*16 + IDX_[a ^ b]] = reorder_sign_(a, b);
    }
  return t;
}
static constexpr CTab make_ctab() {
  CTab t{};
  for (int i = 0; i < 8; ++i)
    for (int j = 0; j < 8; ++j) {
      int a = MASKS_[i], b = MASKS_[j];
      t.tgt[i*8 + j] = IDX_[a ^ b];
      t.sgn[i*8 + j] = reorder_sign_(a, b);
    }
  return t;
}
static constexpr BTab HB = make_btab();
static constexpr CTab HC = make_ctab();
__device__ __constant__ BTab BTAB = HB;
__device__ __constant__ CTab CTAB = HC;
__device__ __constant__ int GRADE8[8] = {0,1,1,1,2,2,2,3};

__device__ __forceinline__ void atomic_fadd(float* p, float v) {
  // Native L2 atomic add, no-return form (STOREcnt tracked; s_endpgm waits idle).
  asm volatile("global_atomic_add_f32 %0, %1, off" :: "v"(p), "v"(v) : "memory");
}

// ---------------------------------------------------------------------------
// Zero-init workspace (sums[N*8] ++ counts[N])
// ---------------------------------------------------------------------------
__global__ void __launch_bounds__(256) ga_zero_kernel(float* __restrict__ p, long long n) {
  long long i = (long long)blockIdx.x * blockDim.x + threadIdx.x;
  if (i < n) p[i] = 0.0f;
}

// ---------------------------------------------------------------------------
// Edge kernel: one wave32 processes tiles of 16 edges.
//   P[16 edges, 16 blades] = W[16 x 64] * C2[64 x 16]
// W[e, i*8+j] = x[dst[e]][i] * x[src[e]][j], contracted with 16 chained
// V_WMMA_F32_16X16X4_F32 ops (K=64 in chunks of 4).
//
// A tile layout (16x4 f32): lanes 0-15 M=0..15 {K0,K1}, lanes 16-31 {K2,K3}.
// B tile layout (4x16 f32): VGPR0 rows {K0(lo),K2(hi)}, VGPR1 rows {K1,K3}.
// D layout (16x16 f32): lane = column N (blade), acc[r] = row M = r + 8*half.
// ---------------------------------------------------------------------------
__global__ void __launch_bounds__(256) ga_edge_kernel(
    const float* __restrict__ x,
    const int* __restrict__ src,
    const int* __restrict__ dst,
    float* __restrict__ sums,
    float* __restrict__ counts,
    long long numTiles)
{
  const int lane = threadIdx.x & 31;
  const int col  = lane & 15;
  const int half = lane >> 4;

  // Build the wave-invariant Cayley B operand once: 16 chunks of 4x16.
  v2f B[16];
  #pragma unroll
  for (int t = 0; t < 16; ++t) {
    int r0 = 4*t + 2*half;
    v2f b;
    b.x = BTAB.v[(r0    )*16 + col];
    b.y = BTAB.v[(r0 + 1)*16 + col];
    B[t] = b;
  }

  const int wavesPerBlock = blockDim.x >> 5;
  long long wave   = (long long)blockIdx.x * wavesPerBlock + (threadIdx.x >> 5);
  long long nWaves = (long long)gridDim.x * wavesPerBlock;

  for (long long tile = wave; tile < numTiles; tile += nWaves) {
    long long e = tile * 16 + col;
    int s = src[e];
    int d = dst[e];

    // Prefetch next tile's index stream (speculative, dropped if OOB).
    __builtin_prefetch(src + e + nWaves * 16, 0, 1);
    __builtin_prefetch(dst + e + nWaves * 16, 0, 1);

    const float4* up = (const float4*)(x + (long long)d * 8); // u = x[dst] (i)
    const float4* vp = (const float4*)(x + (long long)s * 8); // v = x[src] (j)
    float4 u0 = up[0], u1 = up[1];
    float4 w0 = vp[0], w1 = vp[1];
    float U[8] = {u0.x, u0.y, u0.z, u0.w, u1.x, u1.y, u1.z, u1.w};
    // j indices this half-wave touches: half=0 -> {0,1,4,5}; half=1 -> {2,3,6,7}
    float vja = half ? w0.z : w0.x;
    float vjb = half ? w0.w : w0.y;
    float vjc = half ? w1.z : w1.x;
    float vjd = half ? w1.w : w1.y;

    v8f acc = {0.f, 0.f, 0.f, 0.f, 0.f, 0.f, 0.f, 0.f};
    #pragma unroll
    for (int t = 0; t < 16; ++t) {
      float ui = U[t >> 1];
      v2f a;
      if (t & 1) { a.x = ui * vjc; a.y = ui * vjd; }
      else       { a.x = ui * vja; a.y = ui * vjb; }
      acc = __builtin_amdgcn_wmma_f32_16x16x4_f32(
          /*neg_a=*/false, a, /*neg_b=*/false, B[t],
          /*c_mod=*/(short)0, acc, /*reuse_a=*/false, /*reuse_b=*/false);
    }

    // Per-edge neighbor count (one lane per edge).
    if (lane < 16) {
      atomic_fadd(counts + d, 1.0f);
    }
    // Scatter: lane holds blade column 'col' for 8 edge rows.
    if (col < 8) {
      #pragma unroll
      for (int r = 0; r < 8; ++r) {
        int eit = r + 8 * half;                  // edge-in-tile of row M
        int dn  = __shfl(d, eit, 32);            // dst node of that edge
        atomic_fadd(sums + (long long)dn * 8 + col, acc[r]);
      }
    }
  }
}

// Scalar tail for E % 16 edges (not taken for E = 3.2M, kept for generality).
__global__ void ga_edge_tail_kernel(
    const float* __restrict__ x, const int* __restrict__ src,
    const int* __restrict__ dst, float* __restrict__ sums,
    float* __restrict__ counts, long long eBeg, long long eEnd)
{
  long long e = eBeg + (long long)blockIdx.x * blockDim.x + threadIdx.x;
  if (e >= eEnd) return;
  int s = src[e], d = dst[e];
  float u[8], v[8];
  #pragma unroll
  for (int i = 0; i < 8; ++i) { u[i] = x[(long long)d*8 + i]; v[i] = x[(long long)s*8 + i]; }
  float p[8] = {0,0,0,0,0,0,0,0};
  #pragma unroll
  for (int i = 0; i < 8; ++i)
    #pragma unroll
    for (int j = 0; j < 8; ++j)
      p[CTAB.tgt[i*8 + j]] += CTAB.sgn[i*8 + j] * u[i] * v[j];
  #pragma unroll
  for (int k = 0; k < 8; ++k) atomic_fadd(sums + (long long)d*8 + k, p[k]);
  atomic_fadd(counts + d, 1.0f);
}

// ---------------------------------------------------------------------------
// Finalize: mean (isolated-node fallback), grade scaling, bias.
// ---------------------------------------------------------------------------
__global__ void __launch_bounds__(256) ga_finalize_kernel(
    const float* __restrict__ x, const float* __restrict__ sums,
    const float* __restrict__ counts, const float* __restrict__ gw,
    const float* __restrict__ bias, float* __restrict__ out, int n)
{
  int node = blockIdx.x * blockDim.x + threadIdx.x;
  if (node >= n) return;
  float cnt = counts[node];
  bool has = cnt > 0.0f;
  float inv = has ? (1.0f / cnt) : 0.0f;

  float g[8];
  #pragma unroll
  for (int b = 0; b < 8; ++b) g[b] = gw[GRADE8[b]];

  const float4* sp = (const float4*)(sums + (long long)node * 8);
  const float4* xp = (const float4*)(x    + (long long)node * 8);
  float4 s0 = sp[0], s1 = sp[1];
  float4 x0 = xp[0], x1 = xp[1];
  float sv[8] = {s0.x, s0.y, s0.z, s0.w, s1.x, s1.y, s1.z, s1.w};
  float xv[8] = {x0.x, x0.y, x0.z, x0.w, x1.x, x1.y, x1.z, x1.w};
  float ov[8];
  #pragma unroll
  for (int b = 0; b < 8; ++b) {
    float a = has ? sv[b] * inv : xv[b];
    ov[b] = a * g[b] + bias[b];
  }
  float4* op = (float4*)(out + (long long)node * 8);
  op[0] = make_float4(ov[0], ov[1], ov[2], ov[3]);
  op[1] = make_float4(ov[4], ov[5], ov[6], ov[7]);
}

// ---------------------------------------------------------------------------
extern "C" void kernel_launch(void* const* d_in, const int* in_sizes, int n_in,
                              void* d_out, int out_size, void* d_ws, size_t ws_size,
                              hipStream_t stream) {
  const float* x    = (const float*)d_in[0];
  const int*   ei   = (const int*)d_in[1];
  const float* gw   = (const float*)d_in[2];
  const float* bias = (const float*)d_in[3];

  long long N = (long long)in_sizes[0] / 8;
  long long E = (long long)in_sizes[1] / 2;
  const int* src = ei;       // edge_index[0]
  const int* dst = ei + E;   // edge_index[1]

  float* sums   = (float*)d_ws;        // N*8 floats
  float* counts = sums + N * 8;        // N floats
  float* out    = (float*)d_out;

  // 1) zero accumulators
  long long nz = N * 9;
  ga_zero_kernel<<<(int)((nz + 255) / 256), 256, 0, stream>>>(sums, nz);

  // 2) edge scatter (WMMA tiles of 16 edges, 4 tiles per wave via grid-stride)
  long long tiles = E / 16;
  long long rem   = E % 16;
  if (tiles > 0) {
    long long waves  = (tiles + 3) / 4;
    long long blocks = (waves + 7) / 8;       // 8 waves per 256-thread block
    if (blocks < 1) blocks = 1;
    ga_edge_kernel<<<(int)blocks, 256, 0, stream>>>(x, src, dst, sums, counts, tiles);
  }
  if (rem > 0) {
    ga_edge_tail_kernel<<<1, 64, 0, stream>>>(x, src, dst, sums, counts, E - rem, E);
  }

  // 3) finalize
  ga_finalize_kernel<<<(int)((N + 255) / 256), 256, 0, stream>>>(
      x, sums, counts, gw, bias, out, (int)N);
}